// LowRankDynamicConv_32916629357175
// MI455X (gfx1250) — compile-verified
//
#include <hip/hip_runtime.h>

// ---------------------------------------------------------------------------
// LowRankDynamicConv on gfx1250 (MI455X), fp32 WMMA (V_WMMA_F32_16X16X4_F32)
// B=16, T=1024, N=16, C=256, R=32
//
// Pipeline:
//   k_h    : h[256,1024]  = relu([phrase|eos] @ W1 + b1)
//   k_proj : proj[256,8192] = h @ W2 + b2
//   k_z    : zpart[s][b,t,r] = ctx[b,t,ks..] @ proj_chunk   (split-K x8, LDS-staged B)
//   k_zred : z = sum_s zpart[s]
//   k_q    : Q[9,32,256] = tap_j^T @ Wo_slice  (fold conv taps with output proj)
//   k_out  : out = relu(LN(bo + sum_ji z[t+off_ji] @ Q[ji]))
// ---------------------------------------------------------------------------

typedef float v2f __attribute__((ext_vector_type(2)));
typedef float v8f __attribute__((ext_vector_type(8)));

__device__ __forceinline__ v8f wmma_f32(v2f a, v2f b, v8f c) {
    // 8 args: (neg_a, A, neg_b, B, c_mod, C, reuse_a, reuse_b)
    return __builtin_amdgcn_wmma_f32_16x16x4_f32(false, a, false, b,
                                                 (short)0, c, false, false);
}

static constexpr int Bn = 16;
static constexpr int Tn = 1024;
static constexpr int Cn = 256;
static constexpr int Rn = 32;
static constexpr int SPLITK = 8;          // k_z split-K factor

// ---------------------------------------------------------------------------
// Stage 1: h[256,1024] = relu([phrase | eos] @ W1 + b1)
// 16 x 64 tiles = 1024 waves -> 128 blocks x 8 waves
// ---------------------------------------------------------------------------
__global__ __launch_bounds__(256) void k_h(const float* __restrict__ phrase,
                                           const float* __restrict__ eos,
                                           const float* __restrict__ W1,
                                           const float* __restrict__ b1,
                                           float* __restrict__ h) {
    const int lane = threadIdx.x & 31;
    const int wid  = blockIdx.x * (blockDim.x >> 5) + (threadIdx.x >> 5);
    const int nt   = wid & 63;
    const int mt   = wid >> 6;
    const int m    = mt * 16 + (lane & 15);
    const int col  = nt * 16 + (lane & 15);
    const int kh   = (lane >> 4) * 2;

    v8f acc = {};
    for (int kk = 0; kk < 128; ++kk) {
        const int k = kk * 4 + kh;
        v2f a, bb;
        a[0] = (k     < 256) ? phrase[m * 256 + k    ] : eos[k     - 256];
        a[1] = (k + 1 < 256) ? phrase[m * 256 + k + 1] : eos[k + 1 - 256];
        bb[0] = W1[(k    ) * 1024 + col];
        bb[1] = W1[(k + 1) * 1024 + col];
        acc = wmma_f32(a, bb, acc);
    }
    const float bias = b1[col];
#pragma unroll
    for (int v = 0; v < 8; ++v) {
        const int row = mt * 16 + v + (lane >> 4) * 8;
        float x = acc[v] + bias;
        h[row * 1024 + col] = x > 0.f ? x : 0.f;
    }
}

// ---------------------------------------------------------------------------
// Stage 2: proj[256,8192] = h @ W2 + b2
// 16 x 512 tiles = 8192 waves -> 1024 blocks x 8 waves
// ---------------------------------------------------------------------------
__global__ __launch_bounds__(256) void k_proj(const float* __restrict__ h,
                                              const float* __restrict__ W2,
                                              const float* __restrict__ b2,
                                              float* __restrict__ proj) {
    const int lane = threadIdx.x & 31;
    const int wid  = blockIdx.x * (blockDim.x >> 5) + (threadIdx.x >> 5);
    const int nt   = wid & 511;
    const int mt   = wid >> 9;
    const int m    = mt * 16 + (lane & 15);
    const int col  = nt * 16 + (lane & 15);
    const int kh   = (lane >> 4) * 2;

    v8f acc = {};
    for (int kk = 0; kk < 256; ++kk) {
        const int k = kk * 4 + kh;
        v2f a, bb;
        a[0]  = h[m * 1024 + k];
        a[1]  = h[m * 1024 + k + 1];
        bb[0] = W2[(size_t)(k    ) * 8192 + col];
        bb[1] = W2[(size_t)(k + 1) * 8192 + col];
        acc = wmma_f32(a, bb, acc);
    }
    const float bias = b2[col];
#pragma unroll
    for (int v = 0; v < 8; ++v) {
        const int row = mt * 16 + v + (lane >> 4) * 8;
        proj[(size_t)row * 8192 + col] = acc[v] + bias;
    }
}

// ---------------------------------------------------------------------------
// Stage 3 (dominant, HBM-streaming): split-K batched GEMM
//   zpart[s][b,t,r] = sum_{k in chunk s} ctx[b,t,k] * proj[b][k,r]
// Per batch: M=1024, K=4096 -> 8 chunks of 512, N=32.
// Block = 8 waves sharing (b, s); wave owns one 16(t) x 32(r) tile.
// Grid: 16 b x 8 s x 8 t-groups = 1024 blocks x 8 waves = 8192 waves
// (enough in-flight loads to approach 23.3 TB/s).
// The proj chunk for (b, s) is a CONTIGUOUS 64KB region; each 32KB half is
// staged in LDS (row stride padded to 40 floats -> conflict-free ds reads).
// ---------------------------------------------------------------------------
__global__ __launch_bounds__(256) void k_z(const float* __restrict__ ctx,
                                           const float* __restrict__ proj,
                                           float* __restrict__ zpart) {
    __shared__ float sB[256 * 40];        // 256 K-rows x 32 cols, stride 40 (40KB)

    const int lane = threadIdx.x & 31;
    const int w    = threadIdx.x >> 5;
    const int tg   = blockIdx.x & 7;
    const int s    = (blockIdx.x >> 3) & 7;
    const int b    = blockIdx.x >> 6;
    const int tt   = tg * 8 + w;          // t-tile 0..63
    const int mrow = tt * 16 + (lane & 15);
    const int rcol = lane & 15;
    const int kh   = (lane >> 4) * 2;

    const float* __restrict__ ctxb  = ctx + (size_t)b * Tn * 4096;
    // chunk s of proj[b] is contiguous: 512 K-rows * 32 = 16384 floats
    const float* __restrict__ projc = proj + (size_t)b * 16 * 8192 + (size_t)s * 16384;

    v8f acc0 = {}, acc1 = {};
    for (int half = 0; half < 2; ++half) {
        __syncthreads();                  // all waves done reading previous half
        const float4* __restrict__ src4 = (const float4*)(projc + half * 8192);
        for (int j = threadIdx.x; j < 2048; j += 256) {
            const int kl = j >> 3;        // K-row within half (0..255)
            const int rq = j & 7;         // float4 within 32-col row
            *(float4*)&sB[kl * 40 + rq * 4] = src4[j];
        }
        __syncthreads();

        const int kbase = s * 512 + half * 256;
        for (int kk = 0; kk < 64; ++kk) {
            const int k = kk * 4 + kh;    // local K within half
            v2f a, b0, b1;
            a[0] = ctxb[(size_t)mrow * 4096 + kbase + k];
            a[1] = ctxb[(size_t)mrow * 4096 + kbase + k + 1];
            b0[0] = sB[(k    ) * 40 + rcol];
            b0[1] = sB[(k + 1) * 40 + rcol];
            b1[0] = sB[(k    ) * 40 + 16 + rcol];
            b1[1] = sB[(k + 1) * 40 + 16 + rcol];
            acc0 = wmma_f32(a, b0, acc0);
            acc1 = wmma_f32(a, b1, acc1);
        }
    }

    float* __restrict__ zp = zpart + (size_t)s * (Bn * Tn * Rn);
#pragma unroll
    for (int v = 0; v < 8; ++v) {
        const int t = tt * 16 + v + (lane >> 4) * 8;
        zp[((size_t)b * Tn + t) * Rn + rcol]      = acc0[v];
        zp[((size_t)b * Tn + t) * Rn + 16 + rcol] = acc1[v];
    }
}

// ---------------------------------------------------------------------------
// Stage 3b: z = sum_s zpart[s]   (524288 floats = 131072 float4)
// ---------------------------------------------------------------------------
__global__ __launch_bounds__(256) void k_zred(const float* __restrict__ zpart,
                                              float* __restrict__ z) {
    const size_t i = (size_t)blockIdx.x * 256 + threadIdx.x;   // float4 index
    const float4* __restrict__ zp4 = (const float4*)zpart;
    float4 acc = zp4[i];
#pragma unroll
    for (int s = 1; s < SPLITK; ++s) {
        const float4 v = zp4[(size_t)s * 131072 + i];
        acc.x += v.x; acc.y += v.y; acc.z += v.z; acc.w += v.w;
    }
    ((float4*)z)[i] = acc;
}

// ---------------------------------------------------------------------------
// Stage 4: fold conv taps with Wo:  Q[ji,r,e] = sum_d ktap[ji][r][d] * Wo[base+d][e]
// ji enumerates (k,j): k=1 -> 1 tap, k=3 -> 3, k=5 -> 5 (9 total).
// 9 x 2 x 16 tiles = 288 waves = 36 blocks x 8 waves
// ---------------------------------------------------------------------------
__global__ __launch_bounds__(256) void k_q(const float* __restrict__ k1p,
                                           const float* __restrict__ k3p,
                                           const float* __restrict__ k5p,
                                           const float* __restrict__ Wo,
                                           float* __restrict__ Q) {
    const int lane = threadIdx.x & 31;
    const int wid  = blockIdx.x * (blockDim.x >> 5) + (threadIdx.x >> 5);
    const int nt   = wid & 15;
    const int mt   = (wid >> 4) & 1;
    const int ji   = wid >> 5;

    int branch, j;
    if (ji == 0)      { branch = 0; j = 0;      }
    else if (ji < 4)  { branch = 1; j = ji - 1; }
    else              { branch = 2; j = ji - 4; }
    const float* __restrict__ kp = (branch == 0) ? k1p : (branch == 1 ? k3p : k5p);
    const int ksz = (branch == 0) ? 1 : (branch == 1 ? 3 : 5);

    const int r  = mt * 16 + (lane & 15);
    const int e  = nt * 16 + (lane & 15);
    const int kh = (lane >> 4) * 2;

    v8f acc = {};
    for (int kk = 0; kk < 64; ++kk) {
        const int d = kk * 4 + kh;
        v2f a, bb;
        a[0]  = kp[(r * 256 + d    ) * ksz + j];
        a[1]  = kp[(r * 256 + d + 1) * ksz + j];
        bb[0] = Wo[(branch * 256 + d    ) * 256 + e];
        bb[1] = Wo[(branch * 256 + d + 1) * 256 + e];
        acc = wmma_f32(a, bb, acc);
    }
#pragma unroll
    for (int v = 0; v < 8; ++v) {
        const int rr = mt * 16 + v + (lane >> 4) * 8;
        Q[(ji * 32 + rr) * 256 + e] = acc[v];
    }
}

// ---------------------------------------------------------------------------
// Stage 5: out[b,t,e] = relu(LN( bo[e] + sum_ji z[b,t+off_ji,:] @ Q[ji] ))
// One block per (b, t-tile): 512 threads = 16 waves, wave w owns e-tile w.
// ---------------------------------------------------------------------------
__global__ __launch_bounds__(512) void k_out(const float* __restrict__ z,
                                             const float* __restrict__ Q,
                                             const float* __restrict__ bo,
                                             const float* __restrict__ gamma,
                                             const float* __restrict__ beta,
                                             float* __restrict__ out) {
    __shared__ float tile[16 * 256];
    __shared__ float smu[16];
    __shared__ float srs[16];

    const int lane = threadIdx.x & 31;
    const int w    = threadIdx.x >> 5;
    const int tt   = blockIdx.x & 63;
    const int b    = blockIdx.x >> 6;
    const int t0   = tt * 16;
    const int m    = lane & 15;
    const int e    = w * 16 + (lane & 15);
    const int kh   = (lane >> 4) * 2;

    const int off[9] = {0, -1, 0, 1, -2, -1, 0, 1, 2};
    const float* __restrict__ zb = z + (size_t)b * Tn * Rn;

    v8f acc = {};
    for (int ji = 0; ji < 9; ++ji) {
        int tsrc = t0 + m + off[ji];
        const float mask = (tsrc >= 0 && tsrc < Tn) ? 1.f : 0.f;
        tsrc = tsrc < 0 ? 0 : (tsrc > Tn - 1 ? Tn - 1 : tsrc);
        const float* __restrict__ zrow = zb + (size_t)tsrc * Rn;
        const float* __restrict__ Qj   = Q + ji * 32 * 256;
#pragma unroll
        for (int rs = 0; rs < 8; ++rs) {
            const int r = rs * 4 + kh;
            v2f a, bb;
            a[0]  = zrow[r]     * mask;
            a[1]  = zrow[r + 1] * mask;
            bb[0] = Qj[(r    ) * 256 + e];
            bb[1] = Qj[(r + 1) * 256 + e];
            acc = wmma_f32(a, bb, acc);
        }
    }

    const float bias = bo[e];
#pragma unroll
    for (int v = 0; v < 8; ++v) {
        const int row = v + (lane >> 4) * 8;
        tile[row * 256 + e] = acc[v] + bias;
    }
    __syncthreads();

    {   // LayerNorm statistics: wave w reduces row w
        const int row = w;
        float s = 0.f, s2 = 0.f;
#pragma unroll
        for (int i = 0; i < 8; ++i) {
            const float x = tile[row * 256 + lane * 8 + i];
            s += x; s2 += x * x;
        }
#pragma unroll
        for (int d = 16; d >= 1; d >>= 1) {
            s  += __shfl_down(s,  d, 32);
            s2 += __shfl_down(s2, d, 32);
        }
        if (lane == 0) {
            const float mu  = s * (1.f / 256.f);
            const float var = s2 * (1.f / 256.f) - mu * mu;
            smu[row] = mu;
            srs[row] = rsqrtf(var + 1e-5f);
        }
    }
    __syncthreads();

    {   // normalize + affine + relu + store
        const int row = w;
        const float mu = smu[row];
        const float rs = srs[row];
        const size_t base = ((size_t)b * Tn + t0 + row) * Cn;
#pragma unroll
        for (int i = 0; i < 8; ++i) {
            const int ee = lane * 8 + i;
            const float x = tile[row * 256 + ee];
            float y = (x - mu) * rs * gamma[ee] + beta[ee];
            out[base + ee] = y > 0.f ? y : 0.f;
        }
    }
}

// ---------------------------------------------------------------------------
// Launch
// ---------------------------------------------------------------------------
extern "C" void kernel_launch(void* const* d_in, const int* in_sizes, int n_in,
                              void* d_out, int out_size, void* d_ws, size_t ws_size,
                              hipStream_t stream) {
    const float* ctx    = (const float*)d_in[0];
    const float* phrase = (const float*)d_in[1];
    const float* eos    = (const float*)d_in[2];
    const float* W1     = (const float*)d_in[3];
    const float* b1     = (const float*)d_in[4];
    const float* W2     = (const float*)d_in[5];
    const float* b2     = (const float*)d_in[6];
    const float* k1p    = (const float*)d_in[7];
    const float* k3p    = (const float*)d_in[8];
    const float* k5p    = (const float*)d_in[9];
    const float* Wo     = (const float*)d_in[10];
    const float* bo     = (const float*)d_in[11];
    const float* gamma  = (const float*)d_in[12];
    const float* beta   = (const float*)d_in[13];
    float* out = (float*)d_out;

    float* ws    = (float*)d_ws;
    float* h     = ws;                       // 256*1024            =  262144
    float* proj  = ws + 262144;              // 256*8192            = 2097152
    float* zbuf  = ws + 2359296;             // 16*1024*32          =  524288
    float* Qbuf  = ws + 2883584;             // 9*32*256            =   73728
    float* zpart = ws + 2957312;             // 8*16*1024*32        = 4194304
                                             // total 7151616 floats ~ 28.6 MB

    k_h   <<<128,  256, 0, stream>>>(phrase, eos, W1, b1, h);
    k_proj<<<1024, 256, 0, stream>>>(h, W2, b2, proj);
    k_z   <<<1024, 256, 0, stream>>>(ctx, proj, zpart);
    k_zred<<<512,  256, 0, stream>>>(zpart, zbuf);
    k_q   <<<36,   256, 0, stream>>>(k1p, k3p, k5p, Wo, Qbuf);
    k_out <<<1024, 512, 0, stream>>>(zbuf, Qbuf, bo, gamma, beta, out);
}